// YoloDetTarget_83975200571728
// MI455X (gfx1250) — compile-verified
//
#include <hip/hip_runtime.h>

// ---------------------------------------------------------------------------
// YOLO det-target loss: score = max(80 logits), top-k (k = ceil(0.1 N)),
// mask >= 0.25, loss = sum(masked score) + sum(masked boxes).
// Plan: 1 streaming pass (async global->LDS) -> keys(orderable u32)+boxsum
//       + 12-bit histogram; 3-level radix select for exact k-th value;
//       1 digest pass for the masked sum; exact lowest-index tie handling.
// ---------------------------------------------------------------------------

#define ROWF 84
#define CONFT 0.25f
#define TILE 64                 // rows per async tile
#define P1T 64                  // threads per pass1 block
#define CHUNKS_PER_ROW 21       // 84 floats = 21 * float4
#define TILE_CHUNKS (TILE * CHUNKS_PER_ROW)   // 1344
#define TILE_BYTES (TILE_CHUNKS * 16)         // 21504
#define TIE_CAP 1024
#define NB2 1024                // blocks in pass2 / hist_level

struct Ctl {
  unsigned prefix, kRem, cntGreater, uT, includeAll, needTies, tieCount;
  float Tval;
};

__device__ __forceinline__ unsigned f2ord(float f) {
  unsigned b = __float_as_uint(f);
  return b ^ ((b & 0x80000000u) ? 0xFFFFFFFFu : 0x80000000u);
}
__device__ __forceinline__ float ord2f(unsigned u) {
  unsigned b = (u & 0x80000000u) ? (u ^ 0x80000000u) : ~u;
  return __uint_as_float(b);
}

// r points at one 336-byte row (16B aligned): [box float4][20 x logit float4]
__device__ __forceinline__ void row_stats(const float4* r, float& score, float& bs) {
  float4 b = r[0];
  bs = (b.x + b.y) + (b.z + b.w);
  float m = -__builtin_inff();
#pragma unroll
  for (int j = 1; j < 21; ++j) {
    float4 v = r[j];
    m = fmaxf(m, fmaxf(fmaxf(v.x, v.y), fmaxf(v.z, v.w)));
  }
  score = m;
}

__device__ __forceinline__ unsigned lds_off(const void* p) {
  return (unsigned)(unsigned long long)(__attribute__((address_space(3))) const char*)p;
}

// Issue one 64-row tile (21504 B) as 21 async b128 copies per lane.
__device__ __forceinline__ void issue_tile(const float* data, long long maxChunkByte,
                                           int tile, const float4* buf, int tid) {
  unsigned lbase = lds_off(buf);
  long long gb0 = (long long)tile * TILE_BYTES;
#pragma unroll
  for (int j = 0; j < CHUNKS_PER_ROW; ++j) {
    int c = tid + j * P1T;
    long long gb = gb0 + (long long)c * 16;
    if (gb > maxChunkByte) gb = maxChunkByte;  // clamp keeps issue uniform
    const char* g = (const char*)data + gb;
    unsigned l = lbase + (unsigned)c * 16u;
    asm volatile("global_load_async_to_lds_b128 %0, %1, off"
                 :: "v"(l), "v"(g) : "memory");
  }
}

// Pass 1: stream data via async global->LDS double buffer; emit orderable key,
// boxsum, and level-1 (bits 31:20) histogram.
__global__ __launch_bounds__(P1T) void pass1_async(
    const float* __restrict__ data, unsigned* __restrict__ keys,
    float* __restrict__ boxsum, unsigned* __restrict__ hist1, int nRows) {
  __shared__ float4 buf[2][TILE_CHUNKS];   // 2 * 21504 B
  __shared__ unsigned lhist[4096];         // 16 KB
  const int tid = threadIdx.x;
  for (int i = tid; i < 4096; i += P1T) lhist[i] = 0u;
  __syncthreads();

  const int nTiles = (nRows + TILE - 1) / TILE;
  const long long maxChunkByte = (long long)nRows * (ROWF * 4) - 16;
  const int t0 = blockIdx.x;
  const int stride = gridDim.x;

  if (t0 < nTiles) issue_tile(data, maxChunkByte, t0, buf[0], tid);
  int bsel = 0;
  for (int t = t0; t < nTiles; t += stride) {
    const int nxt = t + stride;
    if (nxt < nTiles) {
      issue_tile(data, maxChunkByte, nxt, buf[bsel ^ 1], tid);
      asm volatile("s_wait_asynccnt 21" ::: "memory");  // tile t complete
    } else {
      asm volatile("s_wait_asynccnt 0" ::: "memory");
    }
    __syncthreads();
    const int row = t * TILE + tid;
    if (row < nRows) {
      float score, bs;
      row_stats(&buf[bsel][tid * CHUNKS_PER_ROW], score, bs);
      const unsigned u = f2ord(score);
      keys[row] = u;
      boxsum[row] = bs;
      atomicAdd(&lhist[u >> 20], 1u);
    }
    __syncthreads();
    bsel ^= 1;
  }
  for (int i = tid; i < 4096; i += P1T) {
    unsigned v = lhist[i];
    if (v) atomicAdd(&hist1[i], v);
  }
}

// Fallback level-1 histogram (if workspace too small for digests).
__global__ __launch_bounds__(256) void hist_l1_direct(
    const float* __restrict__ data, unsigned* __restrict__ hist1, int nRows) {
  __shared__ unsigned lhist[4096];
  for (int i = threadIdx.x; i < 4096; i += 256) lhist[i] = 0u;
  __syncthreads();
  for (long long i = blockIdx.x * 256 + threadIdx.x; i < nRows;
       i += (long long)gridDim.x * 256) {
    float s, bs;
    row_stats((const float4*)(data + i * ROWF), s, bs);
    atomicAdd(&lhist[f2ord(s) >> 20], 1u);
  }
  __syncthreads();
  for (int i = threadIdx.x; i < 4096; i += 256) {
    unsigned v = lhist[i];
    if (v) atomicAdd(&hist1[i], v);
  }
}

// Radix levels 2/3: histogram sub-bins of keys matching current prefix.
__global__ __launch_bounds__(256) void hist_level(
    const float* __restrict__ data, const unsigned* __restrict__ keys,
    const Ctl* __restrict__ ctl, unsigned* __restrict__ hist, int nRows,
    int cmpShift, int binShift, unsigned binMask) {
  __shared__ unsigned lhist[4096];
  const int nb = (int)binMask + 1;
  for (int i = threadIdx.x; i < nb; i += 256) lhist[i] = 0u;
  __syncthreads();
  const unsigned pfx = ctl->prefix >> cmpShift;
  for (long long i = blockIdx.x * 256 + threadIdx.x; i < nRows;
       i += (long long)gridDim.x * 256) {
    unsigned u;
    if (keys) {
      u = keys[i];
    } else {
      float s, bs;
      row_stats((const float4*)(data + i * ROWF), s, bs);
      u = f2ord(s);
    }
    if ((u >> cmpShift) == pfx) atomicAdd(&lhist[(u >> binShift) & binMask], 1u);
  }
  __syncthreads();
  for (int i = threadIdx.x; i < nb; i += 256) {
    unsigned v = lhist[i];
    if (v) atomicAdd(&hist[i], v);
  }
}

// Pick the bin containing the k-th largest; update selection state.
__global__ __launch_bounds__(256) void scan_level(
    const unsigned* __restrict__ hist, Ctl* __restrict__ ctl, int bins,
    int binShift, int isFinal, float conf) {
  __shared__ unsigned sh[4096];
  __shared__ unsigned gsum[64];
  const int tid = threadIdx.x;
  for (int i = tid; i < bins; i += 256) sh[i] = hist[i];
  __syncthreads();
  const int gsz = bins / 64;
  if (tid < 64) {
    unsigned s = 0;
    for (int j = 0; j < gsz; ++j) s += sh[tid * gsz + j];
    gsum[tid] = s;
  }
  __syncthreads();
  if (tid == 0) {
    const unsigned kRem = ctl->kRem;
    unsigned cum = 0;
    int g = 63;
    for (; g >= 0; --g) {
      if (cum + gsum[g] >= kRem) break;
      cum += gsum[g];
    }
    if (g < 0) g = 0;
    int b = (g + 1) * gsz - 1;
    for (; b >= g * gsz; --b) {
      if (cum + sh[b] >= kRem) break;
      cum += sh[b];
    }
    if (b < g * gsz) b = g * gsz;
    const unsigned above = cum;
    const unsigned kRem2 = kRem - above;
    const unsigned prefix = ctl->prefix | ((unsigned)b << binShift);
    ctl->cntGreater += above;
    ctl->kRem = kRem2;
    ctl->prefix = prefix;
    if (isFinal) {
      unsigned uT = prefix;
      const float T = ord2f(uT);
      const unsigned cntEq = sh[b];
      unsigned needTies = kRem2;
      unsigned incl = (needTies == cntEq) ? 1u : 0u;
      if (!(T >= conf)) {  // k-th value below CONF: selection = score >= CONF
        uT = f2ord(conf);
        incl = 1u;
        needTies = 0u;
      }
      ctl->uT = uT;
      ctl->includeAll = incl;
      ctl->needTies = needTies;
      ctl->Tval = T;
    }
  }
}

// Masked sum over digests; record tie indices if partial-tie case.
__global__ __launch_bounds__(256) void pass2_partial(
    const float* __restrict__ data, const unsigned* __restrict__ keys,
    const float* __restrict__ boxsum, Ctl* __restrict__ ctl,
    float* __restrict__ partials, unsigned* __restrict__ tieIdx, int nRows) {
  __shared__ float red[256];
  const unsigned uT = ctl->uT;
  const unsigned incl = ctl->includeAll;
  float local = 0.f;
  for (long long i = blockIdx.x * 256 + threadIdx.x; i < nRows;
       i += (long long)gridDim.x * 256) {
    float s, bs;
    unsigned u;
    if (keys) {
      u = keys[i];
      s = ord2f(u);
      bs = boxsum[i];
    } else {
      row_stats((const float4*)(data + i * ROWF), s, bs);
      u = f2ord(s);
    }
    if (u > uT || (u == uT && incl)) {
      local += s + bs;
    } else if (u == uT) {
      unsigned p = atomicAdd(&ctl->tieCount, 1u);
      if (p < TIE_CAP) tieIdx[p] = (unsigned)i;
    }
  }
  red[threadIdx.x] = local;
  __syncthreads();
  for (int o = 128; o > 0; o >>= 1) {
    if (threadIdx.x < o) red[threadIdx.x] += red[threadIdx.x + o];
    __syncthreads();
  }
  if (threadIdx.x == 0) partials[blockIdx.x] = red[0];
}

__global__ __launch_bounds__(256) void finalize_k(
    const float* __restrict__ data, const float* __restrict__ boxsum,
    const Ctl* __restrict__ ctl, const float* __restrict__ partials,
    unsigned* __restrict__ tieIdx, float* __restrict__ out, int nBlocks) {
  __shared__ float red[256];
  const int tid = threadIdx.x;
  float local = 0.f;
  for (int i = tid; i < nBlocks; i += 256) local += partials[i];
  red[tid] = local;
  __syncthreads();
  for (int o = 128; o > 0; o >>= 1) {
    if (tid < o) red[tid] += red[tid + o];
    __syncthreads();
  }
  if (tid == 0) {
    float loss = red[0];
    if (!ctl->includeAll) {
      unsigned cnt = ctl->tieCount;
      if (cnt > TIE_CAP) cnt = TIE_CAP;
      // lowest-index tie-break (matches stable top_k): insertion sort (tiny)
      for (unsigned i = 1; i < cnt; ++i) {
        unsigned v = tieIdx[i];
        int j = (int)i - 1;
        while (j >= 0 && tieIdx[j] > v) { tieIdx[j + 1] = tieIdx[j]; --j; }
        tieIdx[j + 1] = v;
      }
      unsigned take = ctl->needTies;
      if (take > cnt) take = cnt;
      const float T = ctl->Tval;
      for (unsigned i = 0; i < take; ++i) {
        const long long idx = (long long)tieIdx[i];
        float bs;
        if (boxsum) {
          bs = boxsum[idx];
        } else {
          const float* r = data + idx * ROWF;
          bs = (r[0] + r[1]) + (r[2] + r[3]);
        }
        loss += T + bs;
      }
    }
    out[0] = loss;
  }
}

__global__ __launch_bounds__(256) void initk(
    unsigned* hist1, unsigned* hist2, unsigned* hist3, float* partials,
    Ctl* ctl, unsigned k) {
  const int i = blockIdx.x * 256 + threadIdx.x;
  if (i < 4096) { hist1[i] = 0u; hist2[i] = 0u; }
  if (i < 256) hist3[i] = 0u;
  if (i < NB2) partials[i] = 0.f;
  if (i == 0) {
    ctl->prefix = 0u; ctl->kRem = k; ctl->cntGreater = 0u; ctl->uT = 0u;
    ctl->includeAll = 1u; ctl->needTies = 0u; ctl->tieCount = 0u;
    ctl->Tval = 0.f;
  }
}

extern "C" void kernel_launch(void* const* d_in, const int* in_sizes, int n_in,
                              void* d_out, int out_size, void* d_ws, size_t ws_size,
                              hipStream_t stream) {
  const float* data = (const float*)d_in[0];
  const long long total = in_sizes[0];
  const int nRows = (int)(total / ROWF);
  unsigned k = (unsigned)((nRows + 9) / 10);  // ceil(0.1*N) exactly
  if (k < 1u) k = 1u;
  if (k > (unsigned)nRows) k = (unsigned)nRows;

  char* ws = (char*)d_ws;
  size_t off = 0;
  auto take = [&](size_t bytes) -> char* {
    char* p = ws + off;
    off = (off + bytes + 255) & ~(size_t)255;
    return p;
  };
  Ctl* ctl = (Ctl*)take(sizeof(Ctl));
  unsigned* hist1 = (unsigned*)take(4096 * 4);
  unsigned* hist2 = (unsigned*)take(4096 * 4);
  unsigned* hist3 = (unsigned*)take(256 * 4);
  float* partials = (float*)take(NB2 * 4);
  unsigned* tieIdx = (unsigned*)take(TIE_CAP * 4);
  const size_t fixedEnd = off;
  unsigned* keys = nullptr;
  float* boxsum = nullptr;
  if (ws_size >= fixedEnd + (size_t)nRows * 8 + 512) {
    keys = (unsigned*)take((size_t)nRows * 4);
    boxsum = (float*)take((size_t)nRows * 4);
  }

  initk<<<16, 256, 0, stream>>>(hist1, hist2, hist3, partials, ctl, k);
  if (keys) {
    pass1_async<<<4096, P1T, 0, stream>>>(data, keys, boxsum, hist1, nRows);
  } else {
    hist_l1_direct<<<1024, 256, 0, stream>>>(data, hist1, nRows);
  }
  scan_level<<<1, 256, 0, stream>>>(hist1, ctl, 4096, 20, 0, CONFT);
  hist_level<<<NB2, 256, 0, stream>>>(data, keys, ctl, hist2, nRows, 20, 8, 0xFFFu);
  scan_level<<<1, 256, 0, stream>>>(hist2, ctl, 4096, 8, 0, CONFT);
  hist_level<<<NB2, 256, 0, stream>>>(data, keys, ctl, hist3, nRows, 8, 0, 0xFFu);
  scan_level<<<1, 256, 0, stream>>>(hist3, ctl, 256, 0, 1, CONFT);
  pass2_partial<<<NB2, 256, 0, stream>>>(data, keys, boxsum, ctl, partials, tieIdx, nRows);
  finalize_k<<<1, 256, 0, stream>>>(data, boxsum, ctl, partials, tieIdx,
                                    (float*)d_out, NB2);
}